// Model_39676907885426
// MI455X (gfx1250) — compile-verified
//
#include <hip/hip_runtime.h>

// ---------------------------------------------------------------------------
// 100-step chain  t3 = t1 + t2@t3 ; t3 = t1 + t3@t2  (fp32, N=1024),
// then out = t3 + t1.  100 serialized WMMA-f32 GEMM launches; +t1 fused in
// the epilogue, trailing +t1 folded into the last GEMM via alpha=2.
//
// GEMM: 128x128 block tile, BK=32, 8 waves (wave32) tiled 4x2, each wave owns
// 2x4 16x16 accumulators -> 64 v_wmma_f32_16x16x4_f32 per K-tile per wave.
// Double-buffered LDS, register-staged global loads, pair-interleaved B
// layout so every fragment read is one conflict-free ds_load_b64.
// ---------------------------------------------------------------------------

constexpr int NMAT = 1024;

constexpr int BM = 128;
constexpr int BN = 128;
constexpr int BK = 32;
constexpr int KPAIRS = BK / 2;            // 16 K-pairs per tile

constexpr int LDA = BK + 4;               // A tile row stride (floats): 36
                                          // (144 B: 16B-aligned float4 stores,
                                          //  b64 frag reads hit 16 distinct bank pairs)
constexpr int ASZ = BM * LDA;             // A buffer floats
constexpr int BSZ = KPAIRS * BN * 2;      // B buffer floats (pair-interleaved)

typedef __attribute__((ext_vector_type(2))) float v2f;
typedef __attribute__((ext_vector_type(8))) float v8f;

// D = alpha * C + A @ B     (all NMAT x NMAT, row-major, fp32)
__global__ __launch_bounds__(256, 2)
void gemm_wmma_f32(const float* __restrict__ A,
                   const float* __restrict__ B,
                   const float* __restrict__ C,
                   float* __restrict__ D,
                   float alpha)
{
    __shared__ float lA[2][ASZ];          // 2 x 18 KB
    __shared__ float lB[2][BSZ];          // 2 x 16 KB

    const int tid  = threadIdx.x;
    const int lane = tid & 31;            // wave32
    const int wid  = tid >> 5;            // 0..7
    const int half = lane >> 4;           // 0/1 selector per ISA frag layout
    const int l16  = lane & 15;

    const int bm0 = blockIdx.y * BM;
    const int bn0 = blockIdx.x * BN;

    // 8 waves tiled 4(M) x 2(N): each wave computes 32 x 64 = 2x4 16x16 tiles
    const int wm0 = (wid & 3) * 32;
    const int wn0 = (wid >> 2) * 64;

    // ---- staging thread mappings (fixed per thread) ----
    // A: 128 rows x 32 cols, float4/thread, 4 passes of 32 rows
    const int ar = tid >> 3;              // 0..31
    const int ac = (tid & 7) * 4;         // 0,4,...,28
    // B: 16 kpairs x 64 column-pairs, 4 passes; idx -> (p, c)
    //    pass idx = pass*256 + tid ; p = idx>>6 ; c = (idx&63)*2

    v8f acc[2][4];
    {
        const v8f vzero = {};
        #pragma unroll
        for (int mt = 0; mt < 2; ++mt)
            #pragma unroll
            for (int nt = 0; nt < 4; ++nt)
                acc[mt][nt] = vzero;
    }

    float4 ra[4];                         // A staging regs
    float4 rb[4];                         // B staging regs (pair-packed)

    // ---- load K-tile k0 from global into registers ----
    auto load_tile = [&](int k0) {
        #pragma unroll
        for (int p = 0; p < 4; ++p) {
            const int row = ar + p * 32;
            ra[p] = *(const float4*)&A[(size_t)(bm0 + row) * NMAT + k0 + ac];
        }
        #pragma unroll
        for (int p = 0; p < 4; ++p) {
            const int idx = p * 256 + tid;
            const int kp  = idx >> 6;             // 0..15
            const int c   = (idx & 63) * 2;       // 0..126
            const float* b0 = &B[(size_t)(k0 + 2 * kp) * NMAT + bn0 + c];
            const float2 x = *(const float2*)b0;          // row 2p
            const float2 y = *(const float2*)(b0 + NMAT); // row 2p+1
            rb[p].x = x.x; rb[p].y = y.x;                 // (col c  , k pair)
            rb[p].z = x.y; rb[p].w = y.y;                 // (col c+1, k pair)
        }
    };

    // ---- store staged registers into LDS buffer `buf` ----
    auto store_tile = [&](int buf) {
        #pragma unroll
        for (int p = 0; p < 4; ++p) {
            const int row = ar + p * 32;
            *(float4*)&lA[buf][row * LDA + ac] = ra[p];
        }
        #pragma unroll
        for (int p = 0; p < 4; ++p) {
            const int idx = p * 256 + tid;
            const int kp  = idx >> 6;
            const int c   = (idx & 63) * 2;
            *(float4*)&lB[buf][(kp * BN + c) * 2] = rb[p];  // conflict-free b128
        }
    };

    load_tile(0);
    store_tile(0);
    __syncthreads();

    for (int k0 = 0; k0 < NMAT; k0 += BK) {
        const int cur = (k0 / BK) & 1;
        const bool more = (k0 + BK) < NMAT;

        // Issue next tile's global loads before compute: their latency hides
        // under the 64-WMMA chain (all operands are L2-resident).
        if (more) load_tile(k0 + BK);

        #pragma unroll
        for (int kk = 0; kk < BK; kk += 4) {
            // A fragment (16x4): lanes 0-15 hold K=kk,kk+1; lanes 16-31 kk+2,kk+3
            v2f afrag[2];
            #pragma unroll
            for (int mt = 0; mt < 2; ++mt) {
                const float* ap =
                    &lA[cur][(wm0 + mt * 16 + l16) * LDA + kk + 2 * half];
                v2f a; a.x = ap[0]; a.y = ap[1];          // one ds_load_b64
                afrag[mt] = a;
            }
            // B fragment (4x16): pair-interleaved layout -> one ds_load_b64,
            // consecutive lanes at consecutive 8B (no conflicts, no repack).
            v2f bfrag[4];
            #pragma unroll
            for (int nt = 0; nt < 4; ++nt) {
                const float* bp =
                    &lB[cur][(((kk >> 1) + half) * BN +
                              wn0 + nt * 16 + l16) * 2];
                v2f b; b.x = bp[0]; b.y = bp[1];
                bfrag[nt] = b;
            }
            #pragma unroll
            for (int mt = 0; mt < 2; ++mt)
                #pragma unroll
                for (int nt = 0; nt < 4; ++nt)
                    acc[mt][nt] = __builtin_amdgcn_wmma_f32_16x16x4_f32(
                        false, afrag[mt], false, bfrag[nt],
                        (short)0, acc[mt][nt], false, false);
        }

        if (more) store_tile(cur ^ 1);
        __syncthreads();
    }

    // ---- Epilogue: D = alpha*C + acc.  C/D layout: VGPR v holds
    // M = v + 8*half, N = l16; lanes 0-15 / 16-31 write rows r / r+8 ->
    // two 64B coalesced segments per store.
    #pragma unroll
    for (int mt = 0; mt < 2; ++mt) {
        #pragma unroll
        for (int v = 0; v < 8; ++v) {
            const int row = bm0 + wm0 + mt * 16 + half * 8 + v;
            #pragma unroll
            for (int nt = 0; nt < 4; ++nt) {
                const int col = bn0 + wn0 + nt * 16 + l16;
                const size_t idx = (size_t)row * NMAT + col;
                D[idx] = fmaf(alpha, C[idx], acc[mt][nt][v]);
            }
        }
    }
}

extern "C" void kernel_launch(void* const* d_in, const int* in_sizes, int n_in,
                              void* d_out, int out_size, void* d_ws, size_t ws_size,
                              hipStream_t stream)
{
    const float* t1 = (const float*)d_in[0];
    const float* t2 = (const float*)d_in[1];
    float* out  = (float*)d_out;
    float* buf0 = (float*)d_ws;                    // t3 ping
    float* buf1 = buf0 + (size_t)NMAT * NMAT;      // t3 pong (8 MB ws total)

    const dim3 grid(NMAT / BN, NMAT / BM);         // 8 x 8 workgroups
    const dim3 block(256);

    const float* cur = t1;                         // t3 starts as t1
    for (int step = 0; step < 100; ++step) {
        const bool  last  = (step == 99);
        float*      dst   = last ? out : ((step & 1) ? buf1 : buf0);
        const float alpha = last ? 2.0f : 1.0f;    // fold trailing "+ t1"
        if ((step & 1) == 0) {
            gemm_wmma_f32<<<grid, block, 0, stream>>>(t2, cur, t1, dst, alpha);
        } else {
            gemm_wmma_f32<<<grid, block, 0, stream>>>(cur, t2, t1, dst, alpha);
        }
        cur = dst;
    }
}